// MSPNLayer_24378234372318
// MI455X (gfx1250) — compile-verified
//
#include <hip/hip_runtime.h>
#include <cstdint>

typedef __attribute__((ext_vector_type(2))) float v2f;
typedef __attribute__((ext_vector_type(8))) float v8f;

#define B_   2
#define C_   64
#define H_   192
#define W_   192
#define K_   7
#define HW_  (H_*W_)
#define OCH  128          // 2*C
#define KC   68           // padded K-dim for the conv (65 -> 68, zero pad)

// workspace layout (floats)
#define WF_ELEMS   (8 * 17 * 32)          // float2 elements: nt x kt x lane
#define BP_OFF     (WF_ELEMS * 2)         // 8704
#define QK_OFF     9216

// ---------------------------------------------------------------------------
// Kernel 0: fold LayerNorm affine into conv weights, then pack B-fragments.
//   wfold[c][o] = conv_w[o][c]*ln_w[c] (c<64); wfold[64][o]=conv_w[o][64];
//   wfold[65..67][o]=0;  bp[o] = conv_b[o] + sum_c conv_w[o][c]*ln_b[c]
//   Wf[(nt*17+kt)*32 + lane] = { wfold[4kt+2h][nt*16+m], wfold[4kt+2h+1][...] }
// ---------------------------------------------------------------------------
__global__ void __launch_bounds__(128) mspn_prep_kernel(
        const float* __restrict__ conv_w, const float* __restrict__ conv_b,
        const float* __restrict__ ln_w, const float* __restrict__ ln_b,
        float2* __restrict__ Wf, float* __restrict__ bp) {
    __shared__ float wfold[KC * OCH];
    const int o = threadIdx.x;             // 0..127
    float bb = conv_b[o];
    for (int c = 0; c < C_; ++c) {
        float w = conv_w[o * (C_ + 1) + c];
        wfold[c * OCH + o] = w * ln_w[c];
        bb += w * ln_b[c];
    }
    wfold[C_ * OCH + o] = conv_w[o * (C_ + 1) + C_];
    wfold[65 * OCH + o] = 0.f;
    wfold[66 * OCH + o] = 0.f;
    wfold[67 * OCH + o] = 0.f;
    bp[o] = bb;
    __syncthreads();
    for (int t = threadIdx.x; t < 8 * 17 * 32; t += 128) {
        int nt   = t / (17 * 32);
        int rem  = t - nt * (17 * 32);
        int kt   = rem >> 5;
        int lane = rem & 31;
        int mm   = lane & 15, hh = lane >> 4;
        int ca   = 4 * kt + 2 * hh;
        int n    = nt * 16 + mm;
        Wf[t] = make_float2(wfold[ca * OCH + n], wfold[(ca + 1) * OCH + n]);
    }
}

// ---------------------------------------------------------------------------
// Kernel 1: per-pixel LayerNorm stats + 1x1 conv via V_WMMA_F32_16X16X4_F32.
// One wave = 16 pixels (M) x 128 out-channels (8 N-tiles), K = 68 (17 steps).
// B fragments are pre-packed -> one coalesced b64 load per K-step, preloaded
// into registers so all 17 loads clause before the WMMA chain.
// ---------------------------------------------------------------------------
__global__ void __launch_bounds__(256) mspn_qk_kernel(
        const float* __restrict__ g, const float* __restrict__ cd,
        const float* __restrict__ mask, const float2* __restrict__ Wf,
        const float* __restrict__ bp, float* __restrict__ qk) {
    __shared__ float gs[8 * C_ * 17];          // per-wave 64x16 tile, stride 17
    const int lane = threadIdx.x & 31;
    const int wave = threadIdx.x >> 5;
    const int m    = lane & 15;                // pixel within tile
    const int hlf  = lane >> 4;                // K half
    const int tile = blockIdx.x * 8 + wave;
    const int pbase = tile * 16;               // flat pixel over B*H*W
    const int b  = pbase / HW_;
    const int pr = pbase - b * HW_;
    float* gsw = gs + wave * (C_ * 17);

    // ---- stats pass: each lane sums 32 channels of its pixel, stages to LDS
    float s = 0.f, q2 = 0.f;
    const float* gp = g + (size_t)(b * C_ + hlf * 32) * HW_ + pr + m;
    #pragma unroll 4
    for (int i = 0; i < 32; ++i) {
        float v = gp[(size_t)i * HW_];
        s += v; q2 += v * v;
        gsw[(hlf * 32 + i) * 17 + m] = v;
    }
    s  += __shfl_xor(s, 16);
    q2 += __shfl_xor(q2, 16);
    const float mu  = s * (1.f / 64.f);
    const float var = q2 * (1.f / 64.f) - mu * mu;
    const float rs  = rsqrtf(var + 1e-6f);
    const float cdv = cd[b * HW_ + pr + m];
    const float mkv = mask[b * HW_ + pr + m];
    __syncthreads();

    // ---- A fragments in registers: normalized g, then raw cd channel
    v2f afrag[17];
    #pragma unroll
    for (int kt = 0; kt < 16; ++kt) {
        int ca = 4 * kt + 2 * hlf;
        v2f a;
        a.x = (gsw[ca * 17 + m]       - mu) * rs;
        a.y = (gsw[(ca + 1) * 17 + m] - mu) * rs;
        afrag[kt] = a;
    }
    {   // K = 64..67: channel 64 is cd (half 0 only), 65..67 are zero pad
        v2f a; a.x = hlf ? 0.f : cdv; a.y = 0.f;
        afrag[16] = a;
    }

    // ---- 8 N-tiles of 16 out-channels; preload 17 B-frags, then WMMA chain
    for (int nt = 0; nt < 8; ++nt) {
        const int n = nt * 16 + m;
        v2f bfr[17];
        #pragma unroll
        for (int kt = 0; kt < 17; ++kt) {
            float2 w = Wf[(nt * 17 + kt) * 32 + lane];
            v2f bb2; bb2.x = w.x; bb2.y = w.y;
            bfr[kt] = bb2;
        }
        v8f acc = {0.f, 0.f, 0.f, 0.f, 0.f, 0.f, 0.f, 0.f};
        #pragma unroll
        for (int kt = 0; kt < 17; ++kt)
            acc = __builtin_amdgcn_wmma_f32_16x16x4_f32(
                      false, afrag[kt], false, bfr[kt],
                      (short)0, acc, false, false);
        const float bb = bp[n];
        #pragma unroll
        for (int r = 0; r < 8; ++r) {
            int m2 = r + 8 * hlf;                 // D row: vgpr r, lane half
            float val = acc[r] + bb;
            if (nt >= 4) val *= __shfl(mkv, m2);  // mask the k half
            qk[(size_t)(pbase + m2) * OCH + n] = val;
        }
    }
}

// ---------------------------------------------------------------------------
// Kernel 2: 7x7 clamped-window attention via WMMA.
// One wave = 16 consecutive queries in a row. The union of their windows is a
// 7x22 column strip (154 positions); attn[16 q x 154 pos] = q . k is a GEMM
// with K=64 -> 10 N-tiles x 16 v_wmma_f32_16x16x4_f32. Logits round-trip
// through LDS so each lane gets its own 49, then bias+softmax+weighted sums.
// ---------------------------------------------------------------------------
__global__ void __launch_bounds__(32) mspn_attn_kernel(
        const float* __restrict__ cd, const float* __restrict__ sd,
        const float* __restrict__ mask, const float* __restrict__ rpb,
        const float* __restrict__ qk, float* __restrict__ out_cd,
        float* __restrict__ out_mask) {
    __shared__ float ks[160 * 68];             // 154 staged positions, stride 68
    __shared__ float as[16 * 162];             // logit redistribute, stride 162
    const int lane = threadIdx.x;
    const int m   = lane & 15;
    const int hlf = lane >> 4;
    const int tile = blockIdx.x;               // 0 .. B*HW/16-1
    const int b   = tile / (HW_ / 16);
    int rem = tile - b * (HW_ / 16);
    const int h  = rem / (W_ / 16);
    const int w0 = (rem - h * (W_ / 16)) * 16;

    int r0 = h - 3;  r0 = r0 < 0 ? 0 : (r0 > H_ - K_ ? H_ - K_ : r0);
    int a0 = w0 - 3; a0 = a0 < 0 ? 0 : (a0 > W_ - 22 ? W_ - 22 : a0);

    // ---- stage k strip: rows r0..r0+6, cols a0..a0+21, channels 64..127
    const float* kbase = qk + (size_t)(b * HW_ + r0 * W_ + a0) * OCH + C_;
    for (int i = 0; i < 7; ++i) {
        for (int t = lane; t < 22 * 16; t += 32) {
            int j = t >> 4, c4 = t & 15;
            float4 v = *(const float4*)(kbase + (size_t)(i * W_ + j) * OCH + c4 * 4);
            *(float4*)&ks[(i * 22 + j) * 68 + c4 * 4] = v;
        }
    }
    __syncthreads();

    // ---- A fragments: q channels (4kt+2h, +1) of this lane's query pixel
    const int wm = w0 + m;
    const float* qp = qk + (size_t)(b * HW_ + h * W_ + wm) * OCH;
    v2f afr[16];
    #pragma unroll
    for (int kt = 0; kt < 16; ++kt) {
        float2 q2 = *(const float2*)(qp + 4 * kt + 2 * hlf);
        v2f a; a.x = q2.x; a.y = q2.y;
        afr[kt] = a;
    }

    // ---- 10 N-tiles of 16 window positions
    for (int nt = 0; nt < 10; ++nt) {
        const int pos = nt * 16 + m;
        v8f acc = {0.f, 0.f, 0.f, 0.f, 0.f, 0.f, 0.f, 0.f};
        #pragma unroll
        for (int kt = 0; kt < 16; ++kt) {
            float2 k2 = *(const float2*)&ks[pos * 68 + 4 * kt + 2 * hlf];
            v2f bb2; bb2.x = k2.x; bb2.y = k2.y;
            acc = __builtin_amdgcn_wmma_f32_16x16x4_f32(
                      false, afr[kt], false, bb2,
                      (short)0, acc, false, false);
        }
        #pragma unroll
        for (int r = 0; r < 8; ++r)
            as[(r + 8 * hlf) * 162 + pos] = acc[r];
    }
    __syncthreads();

    // ---- per-lane: gather own 49 logits, add relative-position bias
    int cst = wm - 3; cst = cst < 0 ? 0 : (cst > W_ - K_ ? W_ - K_ : cst);
    const int jof  = cst - a0;
    const int rpb0 = 6 - (h - r0);
    const int cpb0 = 6 - (wm - cst);
    float att[49];
    #pragma unroll
    for (int i = 0; i < 7; ++i) {
        #pragma unroll
        for (int j = 0; j < 7; ++j)
            att[i * 7 + j] = as[m * 162 + i * 22 + jof + j]
                           + rpb[(rpb0 + i) * 13 + (cpb0 + j)];
    }

    // ---- softmax over 49
    float mx = att[0];
    #pragma unroll
    for (int t = 1; t < 49; ++t) mx = fmaxf(mx, att[t]);
    float den = 0.f;
    #pragma unroll
    for (int t = 0; t < 49; ++t) {
        float e = __expf(att[t] - mx);
        att[t] = e;
        den += e;
    }
    const float inv = 1.f / den;

    // ---- weighted sums: half 0 aggregates cd window, half 1 aggregates mask
    const float* vsrc = (hlf ? mask : cd) + b * HW_;
    float wsum = 0.f;
    #pragma unroll
    for (int i = 0; i < 7; ++i) {
        #pragma unroll
        for (int j = 0; j < 7; ++j)
            wsum = fmaf(att[i * 7 + j], vsrc[(r0 + i) * W_ + (cst + j)], wsum);
    }
    wsum *= inv;
    float other = __shfl_xor(wsum, 16);        // the other half's aggregate

    if (hlf == 0) {
        const int idx = b * HW_ + h * W_ + wm;
        float mk  = mask[idx];
        float cdq = cd[idx];
        float sdq = sd[idx];
        float cdo = wsum * mk + cdq * (1.f - mk);
        float mo  = other;
        if (sdq > 0.f) { cdo = sdq; mo = mk; }
        out_cd[idx]   = cdo;
        out_mask[idx] = mo;
    }
}

// ---------------------------------------------------------------------------
extern "C" void kernel_launch(void* const* d_in, const int* in_sizes, int n_in,
                              void* d_out, int out_size, void* d_ws, size_t ws_size,
                              hipStream_t stream) {
    (void)in_sizes; (void)n_in; (void)out_size; (void)ws_size;
    const float* g      = (const float*)d_in[0];
    const float* cd     = (const float*)d_in[1];
    const float* sd     = (const float*)d_in[2];
    const float* mask   = (const float*)d_in[3];
    const float* ln_w   = (const float*)d_in[4];
    const float* ln_b   = (const float*)d_in[5];
    const float* conv_w = (const float*)d_in[6];
    const float* conv_b = (const float*)d_in[7];
    const float* rpb    = (const float*)d_in[8];

    float* ws = (float*)d_ws;
    float2* Wf = (float2*)ws;                // 8*17*32 float2 (34.8 KB)
    float* bp  = ws + BP_OFF;                // 128 floats
    float* qk  = ws + QK_OFF;                // B*HW*128 floats (~37.7 MB)

    float* out_cd   = (float*)d_out;
    float* out_mask = out_cd + B_ * HW_;

    mspn_prep_kernel<<<1, 128, 0, stream>>>(conv_w, conv_b, ln_w, ln_b, Wf, bp);
    mspn_qk_kernel<<<(B_ * HW_ / 16) / 8, 256, 0, stream>>>(g, cd, mask, Wf, bp, qk);
    mspn_attn_kernel<<<B_ * HW_ / 16, 32, 0, stream>>>(cd, sd, mask, rpb, qk,
                                                       out_cd, out_mask);
}